// InsuranceAutocorrelationModule_34522947125965
// MI455X (gfx1250) — compile-verified
//
#include <hip/hip_runtime.h>
#include <hip/hip_bf16.h>
#include <math.h>

typedef __attribute__((ext_vector_type(2))) float v2f;
typedef __attribute__((ext_vector_type(8))) float v8f;

#define T_LEN 8192
#define B_N   128
#define NLAG  24
#define FEAT  8

// ---------------------------------------------------------------------------
// Stage 1: one block per batch. Stage seq (feature 0) into LDS with zero pad,
// compute S0/S1 and lag cross-products C[1..24] via FP32 WMMA (Toeplitz trick).
// ws layout per batch: [0]=S0 [1]=S1 [2+(L-1)]=C[L]  (32-float stride)
// ---------------------------------------------------------------------------
__global__ __launch_bounds__(256) void autocorr_stage1(const float* __restrict__ in0,
                                                       float* __restrict__ ws) {
    __shared__ float sx[T_LEN + 64];   // sequence + zero halo (exact edge handling)
    __shared__ float red0[256];
    __shared__ float red1[256];
    __shared__ float cbuf1[8 * 16];    // per-wave lag 0..15 partials
    __shared__ float cbuf2[8 * 16];    // per-wave lag 16..31 partials

    const int b   = blockIdx.x;
    const int tid = threadIdx.x;
    const float* src = in0 + (size_t)b * T_LEN * FEAT;

    // Load feature-0 column once from HBM; accumulate S0/S1 on the fly.
    float s0 = 0.f, s1 = 0.f;
    for (int i = 0; i < T_LEN / 256; ++i) {
        int t = tid + i * 256;
        float v = src[(size_t)t * FEAT];
        sx[t] = v;
        s0 += v;
        s1 += v * v;
    }
    if (tid < 64) sx[T_LEN + tid] = 0.f;   // zero pad -> exact bulk lag sums
    red0[tid] = s0;
    red1[tid] = s1;
    __syncthreads();

    // WMMA phase: wave w covers t in [w*1024, (w+1)*1024).
    // A[i][k] = x[t0+i+k] (16x4), B[k][j] = x[t0+k+j] (4x16): register images of
    // A and B are identical under the ISA 16x4 layout; D[0][j] accumulates
    // sum_t x[t]*x[t+j]. Second accumulator with B shifted +16 covers lags 16..24.
    const int wave = tid >> 5;
    const int lane = tid & 31;
    const int off  = (lane & 15) + 2 * (lane >> 4);
    const int base = wave * (T_LEN / 8);

    v8f d1 = {};
    v8f d2 = {};
#pragma unroll 4
    for (int t0 = base; t0 < base + T_LEN / 8; t0 += 4) {
        const int ia = t0 + off;
        v2f a;  a.x  = sx[ia];      a.y  = sx[ia + 1];
        v2f b2; b2.x = sx[ia + 16]; b2.y = sx[ia + 17];
        d1 = __builtin_amdgcn_wmma_f32_16x16x4_f32(false, a, false, a,
                                                   (short)0, d1, false, false);
        d2 = __builtin_amdgcn_wmma_f32_16x16x4_f32(false, a, false, b2,
                                                   (short)0, d2, false, false);
    }
    // D row M=0 lives in vector element 0 of lanes 0..15.
    if (lane < 16) {
        cbuf1[wave * 16 + lane] = d1[0];
        cbuf2[wave * 16 + lane] = d2[0];
    }
    __syncthreads();

    // Deterministic tree reduction of S0/S1.
    for (int s = 128; s > 0; s >>= 1) {
        if (tid < s) { red0[tid] += red0[tid + s]; red1[tid] += red1[tid + s]; }
        __syncthreads();
    }

    if (tid < NLAG) {
        const int L = tid + 1;
        float c = 0.f;
        if (L < 16) {
            for (int w = 0; w < 8; ++w) c += cbuf1[w * 16 + L];
        } else {
            for (int w = 0; w < 8; ++w) c += cbuf2[w * 16 + (L - 16)];
        }
        ws[b * 32 + 2 + tid] = c;
    }
    if (tid == 0) {
        ws[b * 32 + 0] = red0[0];
        ws[b * 32 + 1] = red1[0];
    }
}

// ---------------------------------------------------------------------------
// Stage 2: one block, one thread per batch. Pearson from sufficient stats
// (affine-invariance makes the seqn normalization a no-op), softmax weights,
// seasonal terms, batch mean.
// ---------------------------------------------------------------------------
__global__ __launch_bounds__(128) void autocorr_stage2(const float* __restrict__ in0,
                                                       const float* __restrict__ lw,
                                                       const float* __restrict__ si,
                                                       const float* __restrict__ ws,
                                                       float* __restrict__ out) {
    __shared__ float red[128];
    const int b = threadIdx.x;
    const float* src = in0 + (size_t)b * T_LEN * FEAT;

    // softmax over the 24 lag weights (cheap, per-thread copy)
    float lwv[NLAG], wv[NLAG];
    float wmax = -1e30f;
    for (int i = 0; i < NLAG; ++i) { lwv[i] = lw[i]; wmax = fmaxf(wmax, lwv[i]); }
    float wsum = 0.f;
    for (int i = 0; i < NLAG; ++i) { wv[i] = __expf(lwv[i] - wmax); wsum += wv[i]; }
    const float inv_wsum = 1.f / wsum;

    const float S0 = ws[b * 32 + 0];
    const float S1 = ws[b * 32 + 1];
    const float si0 = si[0], si1 = si[1];

    float pf0 = 0.f, pf1 = 0.f;   // sum over t <  L   (first 24 elems)
    float sl0 = 0.f, sl1 = 0.f;   // sum over t >= T-L (last 24 elems)
    float acc = 0.f;
    for (int L = 1; L <= NLAG; ++L) {
        const float xf = src[(size_t)(L - 1) * FEAT];
        const float xl = src[(size_t)(T_LEN - L) * FEAT];
        pf0 += xf; pf1 += xf * xf;
        sl0 += xl; sl1 += xl * xl;

        const float N  = (float)(T_LEN - L);
        const float A0 = S0 - sl0, A1 = S1 - sl1;   // window x = seq[:-L]
        const float B0 = S0 - pf0, B1 = S1 - pf1;   // window y = seq[L:]
        const float C  = ws[b * 32 + 2 + (L - 1)];

        const float num = C  - A0 * B0 / N;
        const float dx  = A1 - A0 * A0 / N;
        const float dy  = B1 - B0 * B0 / N;
        float r = num / (sqrtf(dx) * sqrtf(dy));
        r = fminf(1.f, fmaxf(-1.f, r));

        acc += r * (wv[L - 1] * inv_wsum);
        if (L == 12) acc += r * si0;
        if (L == 24) acc += r * si1;
    }
    red[b] = acc;
    __syncthreads();
    for (int s = 64; s > 0; s >>= 1) {
        if (b < s) red[b] += red[b + s];
        __syncthreads();
    }
    if (b == 0) out[0] = red[0] / (float)B_N;
}

extern "C" void kernel_launch(void* const* d_in, const int* in_sizes, int n_in,
                              void* d_out, int out_size, void* d_ws, size_t ws_size,
                              hipStream_t stream) {
    const float* in0 = (const float*)d_in[0];   // input_sequence (B,T,8)
    // d_in[1] (hidden_states) is unused by the reference -> never touched.
    const float* lw  = (const float*)d_in[2];   // lag_weights (24)
    const float* si  = (const float*)d_in[3];   // seasonal_importance (2)
    float* ws  = (float*)d_ws;                  // 128 * 32 floats = 16 KB
    float* out = (float*)d_out;

    autocorr_stage1<<<B_N, 256, 0, stream>>>(in0, ws);
    autocorr_stage2<<<1, 128, 0, stream>>>(in0, lw, si, ws, out);
}